// GraphConv_39840116638415
// MI455X (gfx1250) — compile-verified
//
#include <hip/hip_runtime.h>

#define D_FEAT 128          // D_IN == D_OUT == 128
#define NV4    (D_FEAT / 4) // float4 chunks per feature row

typedef __attribute__((ext_vector_type(2))) float v2f;
typedef __attribute__((ext_vector_type(8))) float v8f;

// ---------------------------------------------------------------------------
// Kernel 1: support = x @ W   via V_WMMA_F32_16X16X4_F32 (fp32 matrix core)
// Block = 128 threads = 4 waves; each block owns a 16-row strip of output,
// wave w covers column tiles {w, w+4} (8 tiles of 16 cols).
// ---------------------------------------------------------------------------
__global__ __launch_bounds__(128) void graphconv_gemm_wmma_f32(
    const float* __restrict__ x, const float* __restrict__ W,
    float* __restrict__ support, int nRows) {
  const int wave = threadIdx.x >> 5;
  const int lane = threadIdx.x & 31;
  const int half = lane >> 4;   // 0: lanes 0-15, 1: lanes 16-31
  const int lm   = lane & 15;
  const int m0   = blockIdx.x * 16;
  if (m0 >= nRows) return;

  // A 16x4 fp32 layout: lane(lm,half) holds A[m0+lm][k + 2*half + {0,1}]
  const float* arow = x + (size_t)(m0 + lm) * D_FEAT + 2 * half;

  for (int nt = wave; nt < D_FEAT / 16; nt += 4) {
    const int n0 = nt * 16;
    // B 4x16 fp32 layout: lane(lm,half) holds W[k + 2*half + {0,1}][n0+lm]
    const float* bptr = W + (size_t)(2 * half) * D_FEAT + n0 + lm;

    v8f acc = {};
#pragma unroll
    for (int k = 0; k < D_FEAT; k += 4) {
      v2f a = *(const v2f*)(arow + k);            // contiguous float2 load
      v2f b;
      b.x = bptr[(size_t)k * D_FEAT];
      b.y = bptr[(size_t)(k + 1) * D_FEAT];
      // D = A*B + C ; emits v_wmma_f32_16x16x4_f32
      acc = __builtin_amdgcn_wmma_f32_16x16x4_f32(
          /*neg_a=*/false, a, /*neg_b=*/false, b,
          /*c_mod=*/(short)0, acc, /*reuse_a=*/false, /*reuse_b=*/false);
    }

    // C/D layout: VGPR r -> row m0 + r + 8*half, column n0 + lm
    float* orow = support + (size_t)(m0 + 8 * half) * D_FEAT + n0 + lm;
#pragma unroll
    for (int r = 0; r < 8; ++r)
      orow[(size_t)r * D_FEAT] = acc[r];
  }
}

// ---------------------------------------------------------------------------
// Kernel 2: out[i] = bias + sum_{e: rows[e]==i} vals[e] * support[cols[e]]
// adj_rows is sorted -> one wave per destination node, binary-search the
// edge range, gather float4 per lane (32 lanes x float4 = 128 cols, 512B
// coalesced per edge, L2-resident since support fits in 192MB L2).
// Deterministic: no atomics, each output row written exactly once.
// ---------------------------------------------------------------------------
__global__ __launch_bounds__(256) void graphconv_spmm_rows(
    const float* __restrict__ support,
    const int* __restrict__ adj_rows, const int* __restrict__ adj_cols,
    const float* __restrict__ adj_vals, const float* __restrict__ bias,
    float* __restrict__ out, int nNodes, int nEdges) {
  const int lane = threadIdx.x & 31;
  const int node = (int)((blockIdx.x * (unsigned)blockDim.x + threadIdx.x) >> 5);
  if (node >= nNodes) return;

  // lower_bound(adj_rows, node)
  int lo = 0, hi = nEdges;
  while (lo < hi) {
    int mid = (lo + hi) >> 1;
    if (adj_rows[mid] < node) lo = mid + 1; else hi = mid;
  }
  const int start = lo;
  // lower_bound(adj_rows, node + 1)
  hi = nEdges;
  while (lo < hi) {
    int mid = (lo + hi) >> 1;
    if (adj_rows[mid] < node + 1) lo = mid + 1; else hi = mid;
  }
  const int end = lo;

  const float4* __restrict__ sup4 = (const float4*)support;
  float4 acc = make_float4(0.f, 0.f, 0.f, 0.f);

  for (int e = start; e < end; ++e) {
    const int   c = adj_cols[e];
    const float v = adj_vals[e];
    const float4 s = sup4[(size_t)c * NV4 + lane];   // global_load_b128
    acc.x += v * s.x; acc.y += v * s.y;
    acc.z += v * s.z; acc.w += v * s.w;
  }

  const float4 b4 = ((const float4*)bias)[lane];
  acc.x += b4.x; acc.y += b4.y; acc.z += b4.z; acc.w += b4.w;
  ((float4*)out)[(size_t)node * NV4 + lane] = acc;   // global_store_b128
}

// ---------------------------------------------------------------------------
extern "C" void kernel_launch(void* const* d_in, const int* in_sizes, int n_in,
                              void* d_out, int out_size, void* d_ws, size_t ws_size,
                              hipStream_t stream) {
  const float* x        = (const float*)d_in[0];
  const float* W        = (const float*)d_in[1];
  const float* bias     = (const float*)d_in[2];
  const int*   adj_rows = (const int*)d_in[3];
  const int*   adj_cols = (const int*)d_in[4];
  const float* adj_vals = (const float*)d_in[5];
  float*       out      = (float*)d_out;

  const int nNodes = in_sizes[0] / D_FEAT;   // 100000
  const int nEdges = in_sizes[3];            // 1600000

  float* support = (float*)d_ws;             // nNodes * 128 floats = 51.2 MB

  // GEMM: one block per 16-row strip
  const int gemmBlocks = (nNodes + 15) / 16;
  graphconv_gemm_wmma_f32<<<gemmBlocks, 128, 0, stream>>>(x, W, support, nNodes);

  // SpMM: one wave per node, 8 waves (256 threads) per block
  const int wavesPerBlock = 256 / 32;
  const int spmmBlocks = (nNodes + wavesPerBlock - 1) / wavesPerBlock;
  graphconv_spmm_rows<<<spmmBlocks, 256, 0, stream>>>(
      support, adj_rows, adj_cols, adj_vals, bias, out, nNodes, nEdges);
}